// ArgumentLocalLogits_87110526697690
// MI455X (gfx1250) — compile-verified
//
#include <hip/hip_runtime.h>

typedef float v2f __attribute__((ext_vector_type(2)));
typedef float v8f __attribute__((ext_vector_type(8)));

#define BS        16
#define CTX_PER   1024
#define ARGS_PER  32
#define KEY_DIM   128
#define D_MODEL   512
#define N_CTX     (BS * CTX_PER)    // 16384
#define N_ARGS    (BS * ARGS_PER)   // 512

// -----------------------------------------------------------------------------
// Kernel 1: keys_all[N_CTX, KEY_DIM] = ctx_values @ W + b   (fp32 WMMA 16x16x4)
// One wave computes a 16(M) x 128(N) strip: 8 accumulators, so the A operand
// (ctx rows, the HBM-resident tensor) is loaded exactly once. W is re-read
// from L2/WGP$ (256 KB, stationary).  1024 waves total.
// -----------------------------------------------------------------------------
__global__ __launch_bounds__(256) void keys_gemm_kernel(
    const float* __restrict__ ctx,   // [N_CTX, D_MODEL]
    const float* __restrict__ W,     // [D_MODEL, KEY_DIM]
    const float* __restrict__ bias,  // [KEY_DIM]
    float* __restrict__ keys)        // [N_CTX, KEY_DIM]
{
    const int wave = (int)((blockIdx.x * blockDim.x + threadIdx.x) >> 5); // 0..1023
    const int lane = (int)(threadIdx.x & 31);
    const int m0 = wave * 16;
    const int mn = lane & 15;        // A: row M ; B/C/D: col N (within a tile)
    const int h  = lane >> 4;        // lane half: K = vgpr + 2*h ; C/D M = v + 8*h

    const float* arow = ctx + (size_t)(m0 + mn) * D_MODEL;

    v8f acc[8];
    #pragma unroll
    for (int t = 0; t < 8; ++t) acc[t] = (v8f){};

    #pragma unroll 2
    for (int k0 = 0; k0 < D_MODEL; k0 += 4) {
        const int ka = k0 + 2 * h;                       // K = vgpr + 2*half
        const v2f a = *(const v2f*)(arow + ka);          // A[M][ka], A[M][ka+1]
        const float* w0 = W + (size_t)ka * KEY_DIM + mn; // B row ka, this lane's N
        #pragma unroll
        for (int t = 0; t < 8; ++t) {
            v2f b;
            b.x = w0[t * 16];                            // B[ka  ][16t+mn]
            b.y = w0[t * 16 + KEY_DIM];                  // B[ka+1][16t+mn]
            acc[t] = __builtin_amdgcn_wmma_f32_16x16x4_f32(
                false, a, false, b, (short)0, acc[t], false, false);
        }
    }

    #pragma unroll
    for (int t = 0; t < 8; ++t) {
        const float bv = bias[t * 16 + mn];
        float* krow = keys + (size_t)(m0 + 8 * h) * KEY_DIM + t * 16 + mn;
        #pragma unroll
        for (int v = 0; v < 8; ++v) {
            // C/D layout: element (M = v + 8*h, N = mn) of tile t
            krow[(size_t)v * KEY_DIM] = acc[t][v] + bv;
        }
    }
}

// -----------------------------------------------------------------------------
// Kernel 2: per segment s: logits_block = Q_s (32x128) @ K_s^T (128x1024).
// One wave computes a 16(M) x 64(N) strip: 4 accumulators, query-tile loads
// reused 4x. keys are L2-resident (8 MB). Also writes rows[p] = p / 1024.
// 512 waves total: seg(16) x mT(2) x nG(16).
// -----------------------------------------------------------------------------
__global__ __launch_bounds__(256) void logits_kernel(
    const float* __restrict__ argv,   // [N_ARGS, KEY_DIM]
    const float* __restrict__ keys,   // [N_CTX, KEY_DIM]
    int*   __restrict__ rows_out,     // [P]
    float* __restrict__ logits_out)   // [P]
{
    const int wave = (int)((blockIdx.x * blockDim.x + threadIdx.x) >> 5); // 0..511
    const int lane = (int)(threadIdx.x & 31);
    const int seg = wave >> 5;        // 0..15
    const int r   = wave & 31;
    const int mT  = r >> 4;           // 0..1
    const int nG  = r & 15;           // 0..15 -> columns nG*64 .. nG*64+63
    const int mn  = lane & 15;
    const int h   = lane >> 4;

    // A: rows of Q_s (M = mn). B[k][n] = K_s[n][k]; per-lane n index = mn,
    // K = vgpr + 2*h -> two consecutive floats of a key row: float2 load.
    const float* qrow = argv + (size_t)(seg * ARGS_PER + mT * 16 + mn) * KEY_DIM;
    const float* kbase = keys + (size_t)(seg * CTX_PER + nG * 64 + mn) * KEY_DIM;

    v8f acc[4];
    #pragma unroll
    for (int t = 0; t < 4; ++t) acc[t] = (v8f){};

    #pragma unroll 4
    for (int k0 = 0; k0 < KEY_DIM; k0 += 4) {
        const int ka = k0 + 2 * h;
        const v2f a = *(const v2f*)(qrow + ka);
        #pragma unroll
        for (int t = 0; t < 4; ++t) {
            const v2f b = *(const v2f*)(kbase + (size_t)(t * 16) * KEY_DIM + ka);
            acc[t] = __builtin_amdgcn_wmma_f32_16x16x4_f32(
                false, a, false, b, (short)0, acc[t], false, false);
        }
    }

    #pragma unroll
    for (int t = 0; t < 4; ++t) {
        const int jcol = nG * 64 + t * 16 + mn;          // column within segment
        #pragma unroll
        for (int v = 0; v < 8; ++v) {
            const int a_global = seg * ARGS_PER + mT * 16 + v + 8 * h;
            const size_t p = (size_t)a_global * CTX_PER + jcol;
            logits_out[p] = acc[t][v];
            rows_out[p]   = a_global;                    // rows[p] = p / CTX_PER
        }
    }
}

// -----------------------------------------------------------------------------
// Launch
// Inputs (setup_inputs order): 0 bs, 1 arg_ids, 2 ctx_ids, 3 arg_values,
//                              4 ctx_values, 5 W, 6 b
// Output: [rows int32 (P)] ++ [logits f32 (P)], P = N_ARGS * CTX_PER
// -----------------------------------------------------------------------------
extern "C" void kernel_launch(void* const* d_in, const int* in_sizes, int n_in,
                              void* d_out, int out_size, void* d_ws, size_t ws_size,
                              hipStream_t stream) {
    (void)in_sizes; (void)n_in; (void)out_size; (void)ws_size;

    const float* arg_values = (const float*)d_in[3];
    const float* ctx_values = (const float*)d_in[4];
    const float* W          = (const float*)d_in[5];
    const float* bias       = (const float*)d_in[6];

    float* keys = (float*)d_ws;                   // N_CTX*KEY_DIM*4 = 8 MB scratch

    const int P = N_ARGS * CTX_PER;               // 524288
    int*   rows_out   = (int*)d_out;
    float* logits_out = (float*)d_out + P;

    // 1024 strip-waves, 8 waves (256 threads) per block -> 128 blocks
    keys_gemm_kernel<<<128, 256, 0, stream>>>(ctx_values, W, bias, keys);
    // 512 strip-waves -> 64 blocks
    logits_kernel<<<64, 256, 0, stream>>>(arg_values, keys, rows_out, logits_out);
}